// CentralAttentiveModule_83416854823697
// MI455X (gfx1250) — compile-verified
//
#include <hip/hip_runtime.h>
#include <hip/hip_bf16.h>

typedef __attribute__((ext_vector_type(2))) float v2f;
typedef __attribute__((ext_vector_type(8))) float v8f;

#define N_PTS   500000
#define DIM     128
#define NCLUS   10000
#define EPSV    1e-5f
#define SLOPE   0.2f

// ---------- order-preserving float <-> uint encoding for atomicMax ----------
__device__ __forceinline__ unsigned enc_f32(float f) {
    unsigned u = __float_as_uint(f);
    return (u & 0x80000000u) ? ~u : (u | 0x80000000u);
}
__device__ __forceinline__ float dec_f32(unsigned e) {
    return (e & 0x80000000u) ? __uint_as_float(e & 0x7FFFFFFFu)
                             : __uint_as_float(~e);
}

// =====================================================================
// Kernel 1: fused triple GEMM  v,k,qp = x @ W^T + b  (fp32 WMMA 16x16x4)
// Block: 256 threads (8 waves), 128 rows of x per block.
// CDNA5 big-LDS: x tile (128x132) + full W matrix (128x132) in LDS
// (135 KB static; WGP supports 320 KB -> 2 blocks/WGP).
// Wave tile = 2 row-tiles x 4 col-tiles: per K-step 6 ds_load_b64 -> 8 wmma
// (0.75 b64/wmma keeps the 64-bank LDS below the XDL issue rate).
// =====================================================================
#define XS_STRIDE 132   // pad 128->132 dwords: conflict-free ds_load_b64
#define ROWS_PB   128

// WSEL: 0 -> store v, 1 -> store k, 2 -> encoded atomicMax into q_enc.
template<int WSEL>
__device__ __forceinline__ void gemm_pass(
    const float* __restrict__ W, const float* __restrict__ bias,
    float* __restrict__ wlds, const float* __restrict__ xs,
    const int* __restrict__ cluster,
    float* __restrict__ outp, unsigned* __restrict__ q_enc,
    int tid, int row0, int rt0, int ctb, int l16, int lh, int n, bool full)
{
    // ---- stage W (128x128) into padded LDS ----
    __syncthreads();   // previous pass finished reading wlds
    for (int u = tid; u < 128 * 32; u += 256) {
        const int r  = u >> 5;
        const int c4 = (u & 31) << 2;
        const float4 val = *(const float4*)(W + (size_t)r * DIM + c4);
        float* dst = &wlds[r * XS_STRIDE + c4];
        dst[0] = val.x; dst[1] = val.y; dst[2] = val.z; dst[3] = val.w;
    }
    __syncthreads();

    v8f acc[2][4];
    #pragma unroll
    for (int i = 0; i < 2; ++i)
        #pragma unroll
        for (int j = 0; j < 4; ++j)
            acc[i][j] = (v8f){0.f,0.f,0.f,0.f,0.f,0.f,0.f,0.f};

    const float* a0 = &xs[((rt0 + 0) * 16 + l16) * XS_STRIDE + 2 * lh];
    const float* a1 = &xs[((rt0 + 1) * 16 + l16) * XS_STRIDE + 2 * lh];
    const float* bb0 = &wlds[((ctb + 0) * 16 + l16) * XS_STRIDE + 2 * lh];
    const float* bb1 = &wlds[((ctb + 1) * 16 + l16) * XS_STRIDE + 2 * lh];
    const float* bb2 = &wlds[((ctb + 2) * 16 + l16) * XS_STRIDE + 2 * lh];
    const float* bb3 = &wlds[((ctb + 3) * 16 + l16) * XS_STRIDE + 2 * lh];

    #pragma unroll 4
    for (int kk = 0; kk < 32; ++kk) {
        const v2f av0 = *(const v2f*)(a0 + kk * 4);   // A: M=l16, K=2*lh+{0,1}
        const v2f av1 = *(const v2f*)(a1 + kk * 4);
        const v2f bv0 = *(const v2f*)(bb0 + kk * 4);  // B[k][n] = W[n][k]
        const v2f bv1 = *(const v2f*)(bb1 + kk * 4);
        const v2f bv2 = *(const v2f*)(bb2 + kk * 4);
        const v2f bv3 = *(const v2f*)(bb3 + kk * 4);
        acc[0][0] = __builtin_amdgcn_wmma_f32_16x16x4_f32(false, av0, false, bv0, (short)0, acc[0][0], false, false);
        acc[0][1] = __builtin_amdgcn_wmma_f32_16x16x4_f32(false, av0, false, bv1, (short)0, acc[0][1], false, false);
        acc[0][2] = __builtin_amdgcn_wmma_f32_16x16x4_f32(false, av0, false, bv2, (short)0, acc[0][2], false, false);
        acc[0][3] = __builtin_amdgcn_wmma_f32_16x16x4_f32(false, av0, false, bv3, (short)0, acc[0][3], false, false);
        acc[1][0] = __builtin_amdgcn_wmma_f32_16x16x4_f32(false, av1, false, bv0, (short)0, acc[1][0], false, false);
        acc[1][1] = __builtin_amdgcn_wmma_f32_16x16x4_f32(false, av1, false, bv1, (short)0, acc[1][1], false, false);
        acc[1][2] = __builtin_amdgcn_wmma_f32_16x16x4_f32(false, av1, false, bv2, (short)0, acc[1][2], false, false);
        acc[1][3] = __builtin_amdgcn_wmma_f32_16x16x4_f32(false, av1, false, bv3, (short)0, acc[1][3], false, false);
    }

    // ---- epilogue ----
    float bb[4];
    #pragma unroll
    for (int j = 0; j < 4; ++j) bb[j] = bias[(ctb + j) * 16 + l16];

    #pragma unroll
    for (int i = 0; i < 2; ++i) {
        const int rbase = row0 + (rt0 + i) * 16 + 8 * lh;  // C layout: M = r + 8*lh
        if (WSEL < 2) {
            if (full) {
                #pragma unroll
                for (int r = 0; r < 8; ++r) {
                    float* rowp = outp + (size_t)(rbase + r) * DIM + l16;
                    #pragma unroll
                    for (int j = 0; j < 4; ++j)
                        rowp[(ctb + j) * 16] = acc[i][j][r] + bb[j];
                }
            } else {
                #pragma unroll
                for (int r = 0; r < 8; ++r) {
                    const int row = rbase + r;
                    if (row < n) {
                        float* rowp = outp + (size_t)row * DIM + l16;
                        #pragma unroll
                        for (int j = 0; j < 4; ++j)
                            rowp[(ctb + j) * 16] = acc[i][j][r] + bb[j];
                    }
                }
            }
        } else {
            #pragma unroll
            for (int r = 0; r < 8; ++r) {
                const int row = rbase + r;
                if (full || row < n) {
                    const int cl = cluster[row];
                    unsigned* qp = q_enc + (size_t)cl * DIM + l16;
                    #pragma unroll
                    for (int j = 0; j < 4; ++j)
                        atomicMax(&qp[(ctb + j) * 16], enc_f32(acc[i][j][r] + bb[j]));
                }
            }
        }
    }
}

__global__ void __launch_bounds__(256)
k_gemm3(const float* __restrict__ x, const int* __restrict__ cluster,
        const float* __restrict__ Wv, const float* __restrict__ bv,
        const float* __restrict__ Wk, const float* __restrict__ bk,
        const float* __restrict__ Wq, const float* __restrict__ bq,
        float* __restrict__ v_out, float* __restrict__ k_out,
        unsigned* __restrict__ q_enc, int n)
{
    __shared__ float xs[ROWS_PB * XS_STRIDE];   // 67.6 KB
    __shared__ float wlds[128 * XS_STRIDE];     // 67.6 KB  (135 KB total)

    const int tid  = threadIdx.x;
    const int row0 = blockIdx.x * ROWS_PB;
    const bool full = (row0 + ROWS_PB <= n);

    // ---- stage x tile (128 x 128) into LDS, clamped at the tail ----
    for (int u = tid; u < ROWS_PB * 32; u += 256) {
        int r  = u >> 5;
        int c4 = (u & 31) << 2;
        int gr = row0 + r; if (gr >= n) gr = n - 1;
        const float4 val = *(const float4*)(x + (size_t)gr * DIM + c4);
        float* dst = &xs[r * XS_STRIDE + c4];
        dst[0] = val.x; dst[1] = val.y; dst[2] = val.z; dst[3] = val.w;
    }

    const int wid  = tid >> 5;
    const int lane = tid & 31;
    const int l16  = lane & 15;
    const int lh   = lane >> 4;

    const int rt0 = (wid >> 1) * 2;   // row tiles rt0, rt0+1  (16 rows each)
    const int ctb = (wid & 1) * 4;    // col tiles ctb .. ctb+3

    gemm_pass<0>(Wv, bv, wlds, xs, cluster, v_out, q_enc, tid, row0, rt0, ctb, l16, lh, n, full);
    gemm_pass<1>(Wk, bk, wlds, xs, cluster, k_out, q_enc, tid, row0, rt0, ctb, l16, lh, n, full);
    gemm_pass<2>(Wq, bq, wlds, xs, cluster, v_out, q_enc, tid, row0, rt0, ctb, l16, lh, n, full);
}

// =====================================================================
// Kernel 2: M[i] = dot(q[cluster[i]], k[i]); per-cluster max via atomicMax
// One wave per point; 8 points per 256-thread block.
// =====================================================================
__global__ void __launch_bounds__(256)
k_dotM(const float* __restrict__ k_arr, const unsigned* __restrict__ q_enc,
       const int* __restrict__ cluster, float* __restrict__ Mv,
       unsigned* __restrict__ m_enc, int n)
{
    const int wid  = threadIdx.x >> 5;
    const int lane = threadIdx.x & 31;
    const int row  = blockIdx.x * 8 + wid;
    if (row >= n) return;

    const int cl = cluster[row];
    const float4 kv = ((const float4*)(k_arr + (size_t)row * DIM))[lane];
    const uint4  qe = ((const uint4*)(q_enc + (size_t)cl * DIM))[lane];
    float p = kv.x * dec_f32(qe.x) + kv.y * dec_f32(qe.y)
            + kv.z * dec_f32(qe.z) + kv.w * dec_f32(qe.w);
    #pragma unroll
    for (int off = 16; off > 0; off >>= 1) p += __shfl_down(p, off, 32);
    if (lane == 0) {
        Mv[row] = p;
        atomicMax(&m_enc[cl], enc_f32(p));
    }
}

// Kernel 3: e = exp(M - m[cluster]); segment sum s += e
__global__ void __launch_bounds__(256)
k_expsum(const float* __restrict__ Mv, const unsigned* __restrict__ m_enc,
         const int* __restrict__ cluster, float* __restrict__ e_arr,
         float* __restrict__ s_sum, int n)
{
    const int row = blockIdx.x * blockDim.x + threadIdx.x;
    if (row >= n) return;
    const int cl = cluster[row];
    const float e = __expf(Mv[row] - dec_f32(m_enc[cl]));
    e_arr[row] = e;
    atomicAdd(&s_sum[cl], e);
}

// Kernel 4: a = e / s[cluster]
__global__ void __launch_bounds__(256)
k_norm(const float* __restrict__ e_arr, const float* __restrict__ s_sum,
       const int* __restrict__ cluster, float* __restrict__ a_arr, int n)
{
    const int row = blockIdx.x * blockDim.x + threadIdx.x;
    if (row >= n) return;
    a_arr[row] = e_arr[row] / s_sum[cluster[row]];
}

// Kernel 5: per-column sum / sumsq of y = a * v   (block-local, then atomics)
__global__ void __launch_bounds__(128)
k_colstats(const float* __restrict__ a_arr, const float* __restrict__ v_arr,
           float* __restrict__ colsum, float* __restrict__ colsumsq, int n)
{
    const int col  = threadIdx.x;        // 0..127
    const int row0 = blockIdx.x * 512;
    float s = 0.f, ss = 0.f;
    for (int i = 0; i < 512; ++i) {
        const int row = row0 + i;
        if (row < n) {
            const float y = a_arr[row] * v_arr[(size_t)row * DIM + col];
            s += y; ss += y * y;
        }
    }
    atomicAdd(&colsum[col], s);
    atomicAdd(&colsumsq[col], ss);
}

// Kernel 6: fold BN into per-column scale/shift
__global__ void __launch_bounds__(128)
k_bnfold(const float* __restrict__ colsum, const float* __restrict__ colsumsq,
         const float* __restrict__ gamma, const float* __restrict__ beta,
         float* __restrict__ sc, float* __restrict__ sh, int n)
{
    const int c = threadIdx.x;
    const float invN = 1.0f / (float)n;
    const float mu  = colsum[c] * invN;
    const float var = colsumsq[c] * invN - mu * mu;
    const float inv = rsqrtf(var + EPSV);
    const float A   = gamma[c] * inv;
    sc[c] = A;
    sh[c] = beta[c] - mu * A;
}

// Kernel 7: out = LeakyReLU( (a*v) * scale + shift )
__global__ void __launch_bounds__(256)
k_finish(const float* __restrict__ a_arr, const float* __restrict__ v_arr,
         const float* __restrict__ sc, const float* __restrict__ sh,
         float* __restrict__ out, int n)
{
    const size_t idx = (size_t)blockIdx.x * blockDim.x + threadIdx.x;
    if (idx >= (size_t)n * DIM) return;
    const int row = (int)(idx >> 7);
    const int col = (int)(idx & 127);
    const float y  = a_arr[row] * v_arr[idx];
    const float yn = y * sc[col] + sh[col];
    out[idx] = (yn >= 0.f) ? yn : SLOPE * yn;
}

// =====================================================================
extern "C" void kernel_launch(void* const* d_in, const int* in_sizes, int n_in,
                              void* d_out, int out_size, void* d_ws, size_t ws_size,
                              hipStream_t stream)
{
    const float* x       = (const float*)d_in[0];
    const int*   cluster = (const int*)  d_in[1];
    const float* Wv      = (const float*)d_in[2];
    const float* bv      = (const float*)d_in[3];
    const float* Wk      = (const float*)d_in[4];
    const float* bk      = (const float*)d_in[5];
    const float* Wq      = (const float*)d_in[6];
    const float* bq      = (const float*)d_in[7];
    const float* gamma   = (const float*)d_in[8];
    const float* beta    = (const float*)d_in[9];
    float* out = (float*)d_out;

    const int n = N_PTS;

    // ---- workspace carve-up ----
    char* p = (char*)d_ws;
    float*    v_arr  = (float*)p;    p += (size_t)n * DIM * sizeof(float);
    float*    k_arr  = (float*)p;    p += (size_t)n * DIM * sizeof(float);
    unsigned* q_enc  = (unsigned*)p; p += (size_t)NCLUS * DIM * sizeof(unsigned);
    unsigned* m_enc  = (unsigned*)p; p += (size_t)NCLUS * sizeof(unsigned);
    float*    s_sum  = (float*)p;    p += (size_t)NCLUS * sizeof(float);
    float*    Mv     = (float*)p;    p += (size_t)n * sizeof(float);
    float*    e_arr  = (float*)p;    p += (size_t)n * sizeof(float);
    float*    a_arr  = (float*)p;    p += (size_t)n * sizeof(float);
    float*    colsum = (float*)p;    p += DIM * sizeof(float);
    float*    colssq = (float*)p;    p += DIM * sizeof(float);
    float*    sc     = (float*)p;    p += DIM * sizeof(float);
    float*    sh     = (float*)p;    p += DIM * sizeof(float);

    // zero reduction buffers (q_enc/m_enc/s_sum contiguous; colsum/colssq contiguous)
    hipMemsetAsync(q_enc, 0,
                   (size_t)NCLUS * DIM * sizeof(unsigned) + 2u * NCLUS * sizeof(unsigned),
                   stream);
    hipMemsetAsync(colsum, 0, 2 * DIM * sizeof(float), stream);

    // 1. fused triple GEMM + segment-max(qp)
    k_gemm3<<<dim3((n + ROWS_PB - 1) / ROWS_PB), dim3(256), 0, stream>>>(
        x, cluster, Wv, bv, Wk, bk, Wq, bq, v_arr, k_arr, q_enc, n);

    // 2. attention logits + per-cluster max
    k_dotM<<<dim3((n + 7) / 8), dim3(256), 0, stream>>>(
        k_arr, q_enc, cluster, Mv, m_enc, n);

    // 3. exp + segment sum
    k_expsum<<<dim3((n + 255) / 256), dim3(256), 0, stream>>>(
        Mv, m_enc, cluster, e_arr, s_sum, n);

    // 4. softmax normalize
    k_norm<<<dim3((n + 255) / 256), dim3(256), 0, stream>>>(
        e_arr, s_sum, cluster, a_arr, n);

    // 5. batch-norm statistics
    k_colstats<<<dim3((n + 511) / 512), dim3(128), 0, stream>>>(
        a_arr, v_arr, colsum, colssq, n);

    // 6. fold BN into scale/shift
    k_bnfold<<<dim3(1), dim3(128), 0, stream>>>(
        colsum, colssq, gamma, beta, sc, sh, n);

    // 7. final elementwise
    k_finish<<<dim3((int)(((size_t)n * DIM + 255) / 256)), dim3(256), 0, stream>>>(
        a_arr, v_arr, sc, sh, out, n);
}